// dwtHaar_2d_43843026158259
// MI455X (gfx1250) — compile-verified
//
#include <hip/hip_runtime.h>
#include <stdint.h>

// Haar DWT, reflect-pad (1,0,1,0), stride 2.
// x: (8,64,512,512) f32  ->  out: [LL|LH|HL|HH] each (8,64,256,256) f32, concatenated.
//
// Memory-bound: 1 GiB total traffic -> ~45us floor at 23.3 TB/s. Strategy:
// CDNA5 async global->LDS staging (ASYNCcnt path), conflict-free ds reads for
// the stride-2 / col-1 access pattern, nontemporal coalesced stores.

#define BC      512            // B*C images
#define H       512
#define W       512
#define HO      256
#define WO      256
#define ROWS    4              // output rows per block
#define IN_ROWS (2 * ROWS)     // input rows staged (8)
#define PLANE   ((size_t)BC * HO * WO)   // 33,554,432 elements per subband

__global__ __launch_bounds__(256) void haar_dwt2d_kernel(
    const float* __restrict__ x, float* __restrict__ out) {

    __shared__ float tile[IN_ROWS * W];   // 8 * 512 * 4B = 16 KB

    const int t   = threadIdx.x;
    const int bid = blockIdx.x;
    const int bc  = bid >> 6;            // image index (64 row-tiles per image)
    const int rb  = bid & 63;            // row-tile index
    const int oi0 = rb * ROWS;           // first output row of this tile
    const int rstart = 2 * oi0 - 1;      // first input row needed (-1 reflects)

    // ---- stage 8 input rows into LDS with CDNA5 async b128 copies ----
    // 8 rows * 512 f32 = 1024 x 16B chunks; 4 chunks per thread, coalesced.
    const uint32_t lds_base = (uint32_t)(uintptr_t)(&tile[0]);
#pragma unroll
    for (int it = 0; it < 4; ++it) {
        const int q   = it * 256 + t;    // chunk id 0..1023
        const int row = q >> 7;          // 128 chunks per input row
        const int c4  = (q & 127) << 2;  // starting float column of chunk
        int gr = rstart + row;
        gr = gr < 0 ? -gr : gr;          // reflect: row -1 -> row 1
        const float* gp = x + (((size_t)bc * H + (size_t)gr) * W + (size_t)c4);
        const uint32_t lds_off = lds_base + (uint32_t)(q << 4);
        asm volatile("global_load_async_to_lds_b128 %0, %1, off"
                     :
                     : "v"(lds_off), "v"((uint64_t)(uintptr_t)gp)
                     : "memory");
    }
    __builtin_amdgcn_s_wait_asynccnt(0);  // my async copies landed in LDS
    __syncthreads();                      // everyone's copies visible

    // ---- compute: thread t -> local output row r = t/64, cols j0 + {0,64,128,192}
    const int r  = t >> 6;
    const int j0 = t & 63;
    // output row oi needs input rows (2*oi-1, 2*oi) == staged rows (2r, 2r+1)
    const float* __restrict__ r0 = &tile[(2 * r    ) * W];
    const float* __restrict__ r1 = &tile[(2 * r + 1) * W];
    const int oi = oi0 + r;
    const size_t obase = ((size_t)bc * HO + (size_t)oi) * WO;
    float* oLL = out +             obase;
    float* oLH = out +     PLANE + obase;
    float* oHL = out + 2 * PLANE + obase;
    float* oHH = out + 3 * PLANE + obase;

#pragma unroll
    for (int k = 0; k < 4; ++k) {
        const int j  = j0 + 64 * k;
        const int c1 = 2 * j;
        const int c0 = (j == 0) ? 1 : (c1 - 1);   // reflect: col -1 -> col 1
        const float a = r0[c0], b = r0[c1];
        const float c = r1[c0], d = r1[c1];
        const float s0 = a + b, s1 = c + d;       // row sums
        const float d0 = b - a, d1 = d - c;       // row diffs
        __builtin_nontemporal_store(0.5f * (s0 + s1), oLL + j);  // LL
        __builtin_nontemporal_store(0.5f * (s0 - s1), oLH + j);  // LH
        __builtin_nontemporal_store(0.5f * (d0 + d1), oHL + j);  // HL
        __builtin_nontemporal_store(0.5f * (d1 - d0), oHH + j);  // HH
    }
}

extern "C" void kernel_launch(void* const* d_in, const int* in_sizes, int n_in,
                              void* d_out, int out_size, void* d_ws, size_t ws_size,
                              hipStream_t stream) {
    (void)in_sizes; (void)n_in; (void)out_size; (void)d_ws; (void)ws_size;
    const float* x  = (const float*)d_in[0];
    float* out      = (float*)d_out;
    const int grid  = BC * (HO / ROWS);   // 512 * 64 = 32768 blocks
    haar_dwt2d_kernel<<<grid, 256, 0, stream>>>(x, out);
}